// GAT_9165460209819
// MI455X (gfx1250) — compile-verified
//
#include <hip/hip_runtime.h>
#include <hip/hip_bf16.h>

typedef __attribute__((ext_vector_type(2))) float v2f;
typedef __attribute__((ext_vector_type(8))) float v8f;
typedef __attribute__((ext_vector_type(4))) unsigned int v4u;
typedef __attribute__((ext_vector_type(8))) int v8i;
typedef __attribute__((ext_vector_type(4))) int v4i;

#define DH 128
#define NEG_SLOPE 0.2f

#if __has_builtin(__builtin_amdgcn_tensor_load_to_lds)
#define USE_TDM 1
#else
#define USE_TDM 0
#endif

// ---------------------------------------------------------------------------
// GEMM: H[n][128] = in[n][128] @ W[128][128], fp32 WMMA 16x16x4.
// One block = 16 rows x 128 cols; 8 waves, wave w owns col tile w (16 cols).
// A tile (16x128 f32 = 8KB) staged in LDS via the Tensor Data Mover (one
// TDM descriptor issued by wave 0, TENSORcnt-tracked); W is L1/L2-resident.
// ---------------------------------------------------------------------------
__global__ __launch_bounds__(256) void gat_gemm_kernel(
    const float* __restrict__ in, const float* __restrict__ W,
    float* __restrict__ H) {
  __shared__ float As[16][DH];  // 8 KB
  const int tid = threadIdx.x;
  const int rowBase = blockIdx.x * 16;
  const int wave = tid >> 5;  // 0..7 -> column tile
  const int lane = tid & 31;

#if USE_TDM
  if (wave == 0) {
    // Build Tensor DMA Descriptor (D#) per CDNA5 ISA §8.3/8.4:
    // 2D tile: tile_dim0=128 elems (row), tile_dim1=16 rows, data_size=4B,
    // tensor_dim0=128, tensor_dim0_stride=128, contiguous into As[16][128].
    const unsigned long long gaddr =
        (unsigned long long)(const void*)(in + (size_t)rowBase * DH);
    const unsigned ldsaddr =
        (unsigned)(unsigned long long)(void*)&As[0][0];  // addr[31:0] = LDS off
    v4u g0;
    g0.x = 1u;               // count=1 (valid), user mode, no gather
    g0.y = ldsaddr;          // lds_addr
    g0.z = (unsigned)gaddr;  // global_addr[31:0]
    g0.w = (unsigned)((gaddr >> 32) & 0x1FFFFFFu) | (2u << 30);  // [56:32]|type=2
    v8i g1;
    g1[0] = (int)(2u << 16);    // workgroup_mask=0, data_size=2 (4 bytes)
    g1[1] = (int)(128u << 16);  // tensor_dim0[15:0]=128 @ bits[63:48]
    g1[2] = (int)(16u << 16);   // tensor_dim0 hi=0 | tensor_dim1[15:0]=16
    g1[3] = (int)(128u << 16);  // tensor_dim1 hi=0 | tile_dim0=128
    g1[4] = 16;                 // tile_dim1=16, tile_dim2=0
    g1[5] = 128;                // tensor_dim0_stride[31:0]=128
    g1[6] = 0;                  // stride hi | tensor_dim1_stride lo (unused 2D)
    g1[7] = 0;
    v4i z = {0, 0, 0, 0};
#if __clang_major__ >= 23
    v8i z8 = {0, 0, 0, 0, 0, 0, 0, 0};
    __builtin_amdgcn_tensor_load_to_lds(g0, g1, z, z, z8, 0);
#else
    __builtin_amdgcn_tensor_load_to_lds(g0, g1, z, z, 0);
#endif
#if __has_builtin(__builtin_amdgcn_s_wait_tensorcnt)
    __builtin_amdgcn_s_wait_tensorcnt(0);
#else
    asm volatile("s_wait_tensorcnt 0x0" ::: "memory");
#endif
  }
#else
  for (int i = tid; i < 16 * DH; i += 256) {
    const int r = i >> 7;
    const int c = i & (DH - 1);
    As[r][c] = in[(size_t)(rowBase + r) * DH + c];
  }
#endif
  __syncthreads();

  const int lhalf = lane & 15;
  const bool hi = lane >= 16;
  const int colBase = wave * 16;
  const int c = colBase + lhalf;
  const int koff = hi ? 2 : 0;  // ISA A-layout: v0 = K0|K2, v1 = K1|K3

  v8f acc = {};
  for (int k0 = 0; k0 < DH; k0 += 4) {
    v2f a, b;
    a.x = As[lhalf][k0 + koff];
    a.y = As[lhalf][k0 + koff + 1];
    b.x = W[(size_t)(k0 + koff) * DH + c];
    b.y = W[(size_t)(k0 + koff + 1) * DH + c];
    acc = __builtin_amdgcn_wmma_f32_16x16x4_f32(
        /*neg_a=*/false, a, /*neg_b=*/false, b,
        /*c_mod=*/(short)0, acc, /*reuse_a=*/false, /*reuse_b=*/false);
  }

  // C/D layout: VGPR j -> row rowBase+j (lanes 0-15) / rowBase+j+8 (16-31)
  const int r0 = rowBase + (hi ? 8 : 0);
#pragma unroll
  for (int j = 0; j < 8; ++j) {
    H[(size_t)(r0 + j) * DH + c] = acc[j];
  }
}

// ---------------------------------------------------------------------------
// Per-node attention scores: s_src[v] = h[v].a_src, s_dst[v] = h[v].a_dst
// One wave32 per node, float4 per lane, shfl reduction.
// ---------------------------------------------------------------------------
__global__ __launch_bounds__(256) void gat_score_kernel(
    const float* __restrict__ H, const float* __restrict__ a_src,
    const float* __restrict__ a_dst, float* __restrict__ s_src,
    float* __restrict__ s_dst, int n) {
  const int wid = (int)(((size_t)blockIdx.x * blockDim.x + threadIdx.x) >> 5);
  const int lane = threadIdx.x & 31;
  if (wid >= n) return;
  const float4 h4 = ((const float4*)(H + (size_t)wid * DH))[lane];
  const float4 as4 = ((const float4*)a_src)[lane];
  const float4 ad4 = ((const float4*)a_dst)[lane];
  float ss = h4.x * as4.x + h4.y * as4.y + h4.z * as4.z + h4.w * as4.w;
  float sd = h4.x * ad4.x + h4.y * ad4.y + h4.z * ad4.z + h4.w * ad4.w;
#pragma unroll
  for (int off = 16; off > 0; off >>= 1) {
    ss += __shfl_down(ss, off, 32);
    sd += __shfl_down(sd, off, 32);
  }
  if (lane == 0) {
    s_src[wid] = ss;
    s_dst[wid] = sd;
  }
}

// ---------------------------------------------------------------------------
// Init: out[v][c] = bias[c]; amax = -inf; denom = 0.
// ---------------------------------------------------------------------------
__global__ __launch_bounds__(256) void gat_init_kernel(
    float* __restrict__ out, const float* __restrict__ b,
    float* __restrict__ amax, float* __restrict__ denom, int n) {
  const size_t i = (size_t)blockIdx.x * blockDim.x + threadIdx.x;
  const size_t total = (size_t)n * DH;
  if (i < total) out[i] = b[i & (DH - 1)];
  if (i < (size_t)n) {
    amax[i] = __int_as_float(0xFF800000);  // -inf
    denom[i] = 0.f;
  }
}

__device__ __forceinline__ float lrelu(float x) {
  return x > 0.f ? x : NEG_SLOPE * x;
}

// Sign-aware float atomic max (amax pre-initialized to -inf).
__device__ __forceinline__ void atomicMaxF(float* addr, float val) {
  if (val >= 0.f)
    atomicMax((int*)addr, __float_as_int(val));
  else
    atomicMin((unsigned int*)addr, (unsigned int)__float_as_int(val));
}

// ---------------------------------------------------------------------------
// Edge pass 1: segment max of leaky_relu(s_src[s] + s_dst[d]) over dst.
// Edge ids [0,E) = real edges, [E, E+n) = self loops.
// ---------------------------------------------------------------------------
__global__ __launch_bounds__(256) void gat_edge_max_kernel(
    const int* __restrict__ src, const int* __restrict__ dst,
    const float* __restrict__ ss, const float* __restrict__ sd,
    float* __restrict__ amax, int E, int n) {
  const int e = (int)((size_t)blockIdx.x * blockDim.x + threadIdx.x);
  if (e >= E + n) return;
  const int s = (e < E) ? src[e] : (e - E);
  const int d = (e < E) ? dst[e] : (e - E);
  atomicMaxF(&amax[d], lrelu(ss[s] + sd[d]));
}

// ---------------------------------------------------------------------------
// Edge pass 2: denom[d] += exp(alpha - amax[d])
// ---------------------------------------------------------------------------
__global__ __launch_bounds__(256) void gat_edge_sum_kernel(
    const int* __restrict__ src, const int* __restrict__ dst,
    const float* __restrict__ ss, const float* __restrict__ sd,
    const float* __restrict__ amax, float* __restrict__ denom, int E, int n) {
  const int e = (int)((size_t)blockIdx.x * blockDim.x + threadIdx.x);
  if (e >= E + n) return;
  const int s = (e < E) ? src[e] : (e - E);
  const int d = (e < E) ? dst[e] : (e - E);
  const float alpha = lrelu(ss[s] + sd[d]);
  atomicAdd(&denom[d], __expf(alpha - amax[d]));
}

// ---------------------------------------------------------------------------
// Edge pass 3: out[d] += coef * h[s]. One wave32 per edge; each lane owns
// 4 consecutive channels (float4 gather, 4 global atomic f32 adds).
// ---------------------------------------------------------------------------
__global__ __launch_bounds__(256) void gat_edge_aggr_kernel(
    const int* __restrict__ src, const int* __restrict__ dst,
    const float* __restrict__ ss, const float* __restrict__ sd,
    const float* __restrict__ amax, const float* __restrict__ denom,
    const float* __restrict__ H, float* __restrict__ out, int E, int n) {
  const int e = (int)(((size_t)blockIdx.x * blockDim.x + threadIdx.x) >> 5);
  const int lane = threadIdx.x & 31;
  if (e >= E + n) return;
  const int s = (e < E) ? src[e] : (e - E);
  const int d = (e < E) ? dst[e] : (e - E);
  const float alpha = lrelu(ss[s] + sd[d]);
  const float coef = __expf(alpha - amax[d]) / denom[d];
  const float4 h4 = ((const float4*)(H + (size_t)s * DH))[lane];
  float* op = out + (size_t)d * DH + lane * 4;
  atomicAdd(op + 0, h4.x * coef);
  atomicAdd(op + 1, h4.y * coef);
  atomicAdd(op + 2, h4.z * coef);
  atomicAdd(op + 3, h4.w * coef);
}

// ---------------------------------------------------------------------------
// In-place ReLU (bias already folded into init).
// ---------------------------------------------------------------------------
__global__ __launch_bounds__(256) void gat_relu_kernel(float* __restrict__ x,
                                                       size_t total) {
  const size_t i = (size_t)blockIdx.x * blockDim.x + threadIdx.x;
  if (i < total) x[i] = fmaxf(x[i], 0.f);
}

// ---------------------------------------------------------------------------
// Host side: 3 GAT layers. ws layout: H | A | s_src | s_dst | amax | denom
// ---------------------------------------------------------------------------
extern "C" void kernel_launch(void* const* d_in, const int* in_sizes, int n_in,
                              void* d_out, int out_size, void* d_ws,
                              size_t ws_size, hipStream_t stream) {
  const float* x = (const float*)d_in[0];
  const int* ei = (const int*)d_in[1];
  const float* Ws = (const float*)d_in[2];
  const float* bs = (const float*)d_in[3];
  const float* a_srcs = (const float*)d_in[4];
  const float* a_dsts = (const float*)d_in[5];

  const int n = in_sizes[0] / DH;  // 100000
  const int E = in_sizes[1] / 2;   // 1600000
  const int* src = ei;
  const int* dstA = ei + E;

  float* H = (float*)d_ws;
  float* A = H + (size_t)n * DH;
  float* ssb = A + (size_t)n * DH;
  float* sdb = ssb + n;
  float* amax = sdb + n;
  float* denom = amax + n;
  float* out = (float*)d_out;

  const int total = E + n;
  const int elemBlocks = (int)(((size_t)n * DH + 255) / 256);
  const int edgeBlocks = (total + 255) / 256;
  const int edgeWaveBlocks = (int)(((size_t)total * 32 + 255) / 256);
  const int nodeWaveBlocks = (int)(((size_t)n * 32 + 255) / 256);

  for (int layer = 0; layer < 3; ++layer) {
    const float* in = (layer == 0) ? x : A;
    float* ob = (layer == 2) ? out : A;
    const float* W = Ws + (size_t)layer * DH * DH;
    const float* b = bs + (size_t)layer * DH;
    const float* av = a_srcs + (size_t)layer * DH;
    const float* dv = a_dsts + (size_t)layer * DH;

    gat_gemm_kernel<<<n / 16, 256, 0, stream>>>(in, W, H);
    gat_score_kernel<<<nodeWaveBlocks, 256, 0, stream>>>(H, av, dv, ssb, sdb,
                                                         n);
    gat_init_kernel<<<elemBlocks, 256, 0, stream>>>(ob, b, amax, denom, n);
    gat_edge_max_kernel<<<edgeBlocks, 256, 0, stream>>>(src, dstA, ssb, sdb,
                                                        amax, E, n);
    gat_edge_sum_kernel<<<edgeBlocks, 256, 0, stream>>>(src, dstA, ssb, sdb,
                                                        amax, denom, E, n);
    gat_edge_aggr_kernel<<<edgeWaveBlocks, 256, 0, stream>>>(
        src, dstA, ssb, sdb, amax, denom, H, ob, E, n);
    gat_relu_kernel<<<elemBlocks, 256, 0, stream>>>(ob, (size_t)n * DH);
  }
}